// GraphFilter_37812892074317
// MI455X (gfx1250) — compile-verified
//
#include <hip/hip_runtime.h>

typedef __attribute__((ext_vector_type(2))) float v2f;
typedef __attribute__((ext_vector_type(8))) float v8f;

#define N_NODES 50000
#define N_EDGES 800000
#define DFEAT   64

// ---------------------------------------------------------------------------
// Dense tap: y[rows,64] (+)= h[rows,64] @ W^T(64x64) + b   using f32 WMMA.
// One wave computes a 16x64 tile: 4 N-tiles x 16 K-steps of 16x16x4 f32 WMMA.
// ---------------------------------------------------------------------------
__global__ __launch_bounds__(256) void gf_gemm_tap(const float* __restrict__ h,
                                                   const float* __restrict__ W,  // W[n][k] row-major 64x64
                                                   const float* __restrict__ b,  // 64
                                                   float* __restrict__ y,
                                                   int accumulate) {
    __shared__ float wlds[64 * 64];

    const int tid = threadIdx.x;
    // Stage W into LDS (4096 floats, float4 per thread x 4 iters)
    {
        const float4* Wv = (const float4*)W;
        float4* Lv = (float4*)wlds;
        #pragma unroll
        for (int i = 0; i < 4; ++i) Lv[tid + i * 256] = Wv[tid + i * 256];
    }
    __syncthreads();

    const int wave = tid >> 5;            // 0..7
    const int lane = tid & 31;
    const int row0 = (blockIdx.x * 8 + wave) * 16;
    const int m    = lane & 15;           // row-in-tile / N-in-tile
    const int half = lane >> 4;           // 0 or 1
    const int c0   = half * 2;            // K sub-column for A/B fragments

    // A fragments: clamp tail rows in-bounds (stores are predicated later).
    int arow = row0 + m;
    if (arow > N_NODES - 1) arow = N_NODES - 1;
    const float* hrow = h + (size_t)arow * DFEAT + c0;

    v2f a[16];
    #pragma unroll
    for (int kt = 0; kt < 16; ++kt)
        a[kt] = *(const v2f*)(hrow + kt * 4);

    // Accumulators init with bias (depends only on N).
    v8f acc[4];
    #pragma unroll
    for (int nt = 0; nt < 4; ++nt) {
        float bias = b[nt * 16 + m];
        #pragma unroll
        for (int r = 0; r < 8; ++r) acc[nt][r] = bias;
    }

    // WMMA sweep: for each N-tile, march K in steps of 4.
    #pragma unroll
    for (int nt = 0; nt < 4; ++nt) {
        const float* wrow = wlds + (nt * 16 + m) * 64 + c0;  // B[kk][n] = W[n][kk]
        #pragma unroll
        for (int kt = 0; kt < 16; ++kt) {
            v2f bb = *(const v2f*)(wrow + kt * 4);
            acc[nt] = __builtin_amdgcn_wmma_f32_16x16x4_f32(
                false, a[kt], false, bb, (short)0, acc[nt], false, false);
        }
    }

    // Store / accumulate D: lane l, vgpr r -> [row0 + r + half*8][nt*16 + m]
    #pragma unroll
    for (int nt = 0; nt < 4; ++nt) {
        #pragma unroll
        for (int r = 0; r < 8; ++r) {
            int row = row0 + r + half * 8;
            if (row < N_NODES) {
                float* p = y + (size_t)row * DFEAT + nt * 16 + m;
                float v = acc[nt][r];
                if (accumulate) v += *p;
                *p = v;
            }
        }
    }
}

// ---------------------------------------------------------------------------
// Graph shift: xout[dst] += xin[src] over all edges. One wave per edge,
// each lane moves 2 contiguous floats (coalesced 256B read per wave),
// scattered with f32 atomics.
// ---------------------------------------------------------------------------
__global__ __launch_bounds__(256) void gf_scatter(const float* __restrict__ xin,
                                                  const int* __restrict__ ei,  // [2][N_EDGES]
                                                  float* __restrict__ xout) {
    int t    = blockIdx.x * 256 + threadIdx.x;
    int e    = t >> 5;
    int lane = t & 31;
    if (e >= N_EDGES) return;
    int src = ei[e];
    int dst = ei[N_EDGES + e];
    const float2 v = *(const float2*)(xin + (size_t)src * DFEAT + lane * 2);
    float* p = xout + (size_t)dst * DFEAT + lane * 2;
    atomicAdd(p,     v.x);
    atomicAdd(p + 1, v.y);
}

__global__ __launch_bounds__(256) void gf_zero(float4* __restrict__ p, int n4) {
    int i = blockIdx.x * 256 + threadIdx.x;
    if (i < n4) p[i] = make_float4(0.f, 0.f, 0.f, 0.f);
}

extern "C" void kernel_launch(void* const* d_in, const int* in_sizes, int n_in,
                              void* d_out, int out_size, void* d_ws, size_t ws_size,
                              hipStream_t stream) {
    const float* x  = (const float*)d_in[0];
    const int*   ei = (const int*)d_in[1];
    const float* W  = (const float*)d_in[2];   // [4][64][64]
    const float* b  = (const float*)d_in[3];   // [4][64]
    float* y = (float*)d_out;

    float* t0 = (float*)d_ws;                          // 50000*64 f32 = 12.8 MB
    float* t1 = t0 + (size_t)N_NODES * DFEAT;          // second ping-pong buffer

    const int gemm_blocks  = (N_NODES + 127) / 128;            // 8 waves x 16 rows
    const int scat_blocks  = (N_EDGES * 32 + 255) / 256;       // one wave per edge
    const int n4           = N_NODES * DFEAT / 4;
    const int zero_blocks  = (n4 + 255) / 256;

    // tap 0: y = x @ W0^T + b0
    gf_gemm_tap<<<gemm_blocks, 256, 0, stream>>>(x, W, b, y, 0);

    // tap 1
    gf_zero<<<zero_blocks, 256, 0, stream>>>((float4*)t0, n4);
    gf_scatter<<<scat_blocks, 256, 0, stream>>>(x, ei, t0);
    gf_gemm_tap<<<gemm_blocks, 256, 0, stream>>>(t0, W + 4096, b + 64, y, 1);

    // tap 2
    gf_zero<<<zero_blocks, 256, 0, stream>>>((float4*)t1, n4);
    gf_scatter<<<scat_blocks, 256, 0, stream>>>(t0, ei, t1);
    gf_gemm_tap<<<gemm_blocks, 256, 0, stream>>>(t1, W + 2 * 4096, b + 2 * 64, y, 1);

    // tap 3
    gf_zero<<<zero_blocks, 256, 0, stream>>>((float4*)t0, n4);
    gf_scatter<<<scat_blocks, 256, 0, stream>>>(t1, ei, t0);
    gf_gemm_tap<<<gemm_blocks, 256, 0, stream>>>(t0, W + 3 * 4096, b + 3 * 64, y, 1);
}